// Item_GraphConvolution_mid_16140487098643
// MI455X (gfx1250) — compile-verified
//
#include <hip/hip_runtime.h>

// Problem constants from the reference
#define F_DIM 512
#define D_DIM 128
#define KPANEL 64

typedef __attribute__((ext_vector_type(2))) float v2f;
typedef __attribute__((ext_vector_type(4))) float f4;
typedef __attribute__((ext_vector_type(8))) float v8f;

// ---------------------------------------------------------------------------
// Kernel 1: support = relu(feature @ W), fp32 WMMA 16x16x4.
// Block = 256 threads (8 wave32). Block tile = 16 rows x 128 cols (all of D).
// Wave w computes the 16x16 tile at columns [16w, 16w+16).
// K panel of 64 staged in LDS (A: 16x64 pad 68, B: 64x128 pad 132).
// ---------------------------------------------------------------------------
__global__ __launch_bounds__(256) void gcn_gemm_relu(
    const float* __restrict__ feature,   // [N, 512] row-major
    const float* __restrict__ W,         // [512, 128] row-major
    float* __restrict__ support,         // [N, 128]
    int N) {
  __shared__ float As[16 * 68];        // 16 rows x KPANEL (stride 68)
  __shared__ float Bs[KPANEL * 132];   // KPANEL rows x 128 cols (stride 132)

  const int t       = threadIdx.x;
  const int wave    = t >> 5;
  const int lane    = t & 31;
  const int half    = lane >> 4;       // 0: K=0,1 | 1: K=2,3 (A frag layout)
  const int l16     = lane & 15;
  const int rowBase = blockIdx.x * 16;
  const int colBase = wave * 16;

  v8f acc = {};

  // A-panel staging assignment: thread t loads one float4 of the 16x64 panel.
  const int aRow = t >> 4;            // 0..15
  const int aC4  = (t & 15) * 4;      // 0..60
  const int aRowG = min(rowBase + aRow, N - 1);
  const long aBase = (long)aRowG * F_DIM + aC4;

  for (int k0 = 0; k0 < F_DIM; k0 += KPANEL) {
    // ---- stage A (16x64 fp32, one b128 load per thread) ----
    *(f4*)&As[aRow * 68 + aC4] = *(const f4*)&feature[aBase + k0];
    if (k0 + KPANEL < F_DIM) {
      __builtin_prefetch(&feature[aBase + k0 + KPANEL], 0, 1);
    }
    // ---- stage B = W panel (64x128 fp32, 8 b128 loads per thread) ----
#pragma unroll
    for (int i = 0; i < 8; ++i) {
      int f  = t + i * 256;           // float4 index within panel, 0..2047
      int br = f >> 5;                // panel row 0..63
      int bc = (f & 31) * 4;          // col 0..124
      *(f4*)&Bs[br * 132 + bc] = *(const f4*)&W[(k0 + br) * D_DIM + bc];
    }
    __syncthreads();

    // ---- 16 WMMA steps over the K panel ----
#pragma unroll
    for (int kk = 0; kk < KPANEL; kk += 4) {
      const int ka = kk + half * 2;
      v2f a, b;
      // A 16x4 f32 layout: lanes 0-15 M=lane, VGPR0=K0 VGPR1=K1;
      //                    lanes 16-31 M=lane-16, VGPR0=K2 VGPR1=K3.
      a.x = As[l16 * 68 + ka];
      a.y = As[l16 * 68 + ka + 1];
      // B 4x16 f32 (K x N): mirror layout, lanes hold N=l16.
      b.x = Bs[ka * 132 + colBase + l16];
      b.y = Bs[(ka + 1) * 132 + colBase + l16];
      acc = __builtin_amdgcn_wmma_f32_16x16x4_f32(
          false, a, false, b, (short)0, acc, false, false);
    }
    __syncthreads();
  }

  // ---- ReLU + store. C/D layout: VGPR i -> M = i + 8*half, N = l16. ----
#pragma unroll
  for (int i = 0; i < 8; ++i) {
    float v = acc[i] > 0.0f ? acc[i] : 0.0f;
    int r = rowBase + i + half * 8;
    if (r < N) support[(long)r * D_DIM + colBase + l16] = v;
  }
}

// ---------------------------------------------------------------------------
// Kernel 2: out = support + bias  (identity term of adj_low, plus bias).
// ---------------------------------------------------------------------------
__global__ __launch_bounds__(256) void gcn_init_out(
    const float* __restrict__ support,
    const float* __restrict__ bias,
    float* __restrict__ out,
    long total) {
  long i = (long)blockIdx.x * blockDim.x + threadIdx.x;
  if (i < total) out[i] = support[i] + bias[i & (D_DIM - 1)];
}

// ---------------------------------------------------------------------------
// Kernel 3: edge scatter-add.  One wave per edge; lane handles dims
// {lane, lane+32, lane+64, lane+96} -> coalesced gathers + f32 atomics.
// ---------------------------------------------------------------------------
__global__ __launch_bounds__(256) void gcn_edge_scatter(
    const float* __restrict__ support,
    const float* __restrict__ edge_vals,
    const int* __restrict__ edge_index,   // [2, E]: row = [0:E), col = [E:2E)
    float* __restrict__ out,
    int E) {
  long gid = (long)blockIdx.x * blockDim.x + threadIdx.x;
  int e    = (int)(gid >> 5);
  int lane = (int)(gid & 31);
  if (e >= E) return;

  int dst  = edge_index[e];       // row (destination of segment_sum)
  int src  = edge_index[E + e];   // col (gather source)
  float v  = edge_vals[e];

  const float* sp = support + (long)src * D_DIM;
  float*       op = out     + (long)dst * D_DIM;
#pragma unroll
  for (int j = 0; j < 4; ++j) {
    int d = lane + 32 * j;
    atomicAdd(&op[d], v * sp[d]);
  }
}

// ---------------------------------------------------------------------------
extern "C" void kernel_launch(void* const* d_in, const int* in_sizes, int n_in,
                              void* d_out, int out_size, void* d_ws, size_t ws_size,
                              hipStream_t stream) {
  const float* feature    = (const float*)d_in[0];
  const float* W          = (const float*)d_in[1];
  const float* bias       = (const float*)d_in[2];
  const float* edge_vals  = (const float*)d_in[3];
  const int*   edge_index = (const int*)d_in[4];
  float*       out        = (float*)d_out;

  const int N = in_sizes[0] / F_DIM;     // 100000
  const int E = in_sizes[3];             // 1600000

  float* support = (float*)d_ws;         // N * 128 floats of scratch

  // 1) support = relu(feature @ W)
  dim3 gemmGrid((N + 15) / 16);
  gcn_gemm_relu<<<gemmGrid, 256, 0, stream>>>(feature, W, support, N);

  // 2) out = support + bias
  long total = (long)N * D_DIM;
  dim3 initGrid((unsigned)((total + 255) / 256));
  gcn_init_out<<<initGrid, 256, 0, stream>>>(support, bias, out, total);

  // 3) out[row] += edge_vals * support[col]
  long lanesNeeded = (long)E * 32;
  dim3 scatGrid((unsigned)((lanesNeeded + 255) / 256));
  gcn_edge_scatter<<<scatGrid, 256, 0, stream>>>(support, edge_vals, edge_index, out, E);
}